// DualLoss_32581621908078
// MI455X (gfx1250) — compile-verified
//
#include <hip/hip_runtime.h>
#include <hip/hip_bf16.h>

// Problem constants (from reference setup_inputs)
constexpr int kB = 16;
constexpr int kN = 2048;
constexpr int kM = 16;
constexpr int kS = 128;
constexpr float kInfClamp = 1e30f;
constexpr float kFourPi = 12.566370614359172f;

typedef float v2f __attribute__((ext_vector_type(2)));
typedef float v8f __attribute__((ext_vector_type(8)));

// ---------------------------------------------------------------------------
// Kernel 1: per-(b,m) pairwise squared distances via V_WMMA_F32_16X16X4_F32,
// producing d1[b][n][m] = min_s dist and d2[b][m][s] = min_n dist.
// Block = 256 threads (8 waves). Wave w owns N-tiles [16w, 16w+16) and loops
// all 8 S-row tiles per N-tile.
// ---------------------------------------------------------------------------
__global__ __launch_bounds__(256) void dist_min_kernel(
    const float* __restrict__ pcl,    // (B,N,M,3)
    const float* __restrict__ prim,   // (B,M,S,3)
    float* __restrict__ d1,           // (B,N,M)
    float* __restrict__ d2)           // (B,M,S)
{
    __shared__ float rowpart[8 * kS];   // per-wave partial row minima (4 KB)

    const int b    = blockIdx.x / kM;
    const int m    = blockIdx.x % kM;
    const int tid  = threadIdx.x;
    const int w    = tid >> 5;          // wave id 0..7
    const int lane = tid & 31;
    const int l16  = lane & 15;
    const bool hi  = (lane >= 16);

    // ---- Load A tiles: 8 tiles of 16x4 f32 (rows = primitive points) ----
    // ISA layout (32-bit A 16x4): lanes 0-15 hold K=0,1 of row M=lane;
    // lanes 16-31 hold K=2,3 of row M=lane-16.
    // A row for point p: (-2px, -2py, -2pz, |p|^2)
    v2f aReg[8];
#pragma unroll
    for (int r = 0; r < 8; ++r) {
        const int s = 16 * r + l16;
        const float* pp = prim + (((size_t)(b * kM + m)) * kS + s) * 3;
        const float px = pp[0], py = pp[1], pz = pp[2];
        const float pn2 = px * px + py * py + pz * pz;
        v2f a;
        a.x = hi ? (-2.0f * pz) : (-2.0f * px);
        a.y = hi ? pn2          : (-2.0f * py);
        aReg[r] = a;
    }

    // Running row minima: rowacc[r][v] covers row s = 16r + v + (hi?8:0)
    // over this wave's 256 columns.
    float rowacc[8][8];
#pragma unroll
    for (int r = 0; r < 8; ++r)
#pragma unroll
        for (int v = 0; v < 8; ++v)
            rowacc[r][v] = 3.0e38f;

    for (int tt = 0; tt < 16; ++tt) {
        const int t = w * 16 + tt;          // N-tile index (0..127)
        const int n = 16 * t + l16;         // this lane's column
        const float* qp = pcl + (((size_t)(b * kN + n)) * kM + m) * 3;
        const float qx = qp[0], qy = qp[1], qz = qp[2];
        const float qn2 = qx * qx + qy * qy + qz * qz;

        // B 4x16 f32 layout: VGPR0 lanes0-15 -> K=0 (qx), lanes16-31 -> K=2 (qz);
        //                    VGPR1 lanes0-15 -> K=1 (qy), lanes16-31 -> K=3 (1).
        v2f bv;
        bv.x = hi ? qz   : qx;
        bv.y = hi ? 1.0f : qy;

        float cmin = 3.4e38f;
#pragma unroll
        for (int r = 0; r < 8; ++r) {
            v8f c = {0.f, 0.f, 0.f, 0.f, 0.f, 0.f, 0.f, 0.f};
            // D = A x B + C  ->  |p|^2 - 2 p.q  (per 16x16 tile)
            c = __builtin_amdgcn_wmma_f32_16x16x4_f32(
                /*neg_a=*/false, aReg[r], /*neg_b=*/false, bv,
                /*c_mod=*/(short)0, c, /*reuse_a=*/false, /*reuse_b=*/false);
#pragma unroll
            for (int v = 0; v < 8; ++v) {
                const float dval = c[v] + qn2;        // + |q|^2 (column-const per lane)
                rowacc[r][v] = fminf(rowacc[r][v], dval);
                cmin = fminf(cmin, dval);
            }
        }
        // Column min: lanes L and L^16 see the same column n (rows 0-7 vs 8-15
        // of each tile) -> combine halves.
        cmin = fminf(cmin, __shfl_xor(cmin, 16, 32));
        if (!hi) {
            d1[((size_t)(b * kN + n)) * kM + m] = cmin;   // exclusive per wave
        }
    }

    // ---- Row minima: reduce across the 16 lanes of each half-wave ----
#pragma unroll
    for (int r = 0; r < 8; ++r) {
#pragma unroll
        for (int v = 0; v < 8; ++v) {
            float rm = rowacc[r][v];
            rm = fminf(rm, __shfl_xor(rm, 1, 32));
            rm = fminf(rm, __shfl_xor(rm, 2, 32));
            rm = fminf(rm, __shfl_xor(rm, 4, 32));
            rm = fminf(rm, __shfl_xor(rm, 8, 32));
            if (l16 == 0) {
                const int s = 16 * r + v + (hi ? 8 : 0);
                rowpart[w * kS + s] = rm;
            }
        }
    }
    __syncthreads();

    if (tid < kS) {
        float rm = rowpart[tid];
#pragma unroll
        for (int ww = 1; ww < 8; ++ww)
            rm = fminf(rm, rowpart[ww * kS + tid]);
        d2[((size_t)(b * kM + m)) * kS + tid] = rm;
    }
}

// ---------------------------------------------------------------------------
// Kernel 2: pcl_to_prim. One thread per (b,n): sort the M=16 distances with a
// fixed odd-even transposition network (stays in registers), stick-breaking
// cumulative product, block-reduce to deterministic per-block partials.
// ---------------------------------------------------------------------------
__global__ __launch_bounds__(256) void pcl_to_prim_kernel(
    const float* __restrict__ d1,     // (B,N,M)
    const float* __restrict__ probs,  // (B,M)
    float* __restrict__ partials)     // (gridDim.x)
{
    __shared__ float red[256];
    const int i = blockIdx.x * 256 + threadIdx.x;   // < B*N
    const int b = i / kN;
    const int n = i % kN;

    float d[kM], p[kM];
#pragma unroll
    for (int mm = 0; mm < kM; ++mm) {
        d[mm] = d1[((size_t)(b * kN + n)) * kM + mm];
        p[mm] = probs[b * kM + mm];
    }

    // Odd-even transposition sort, 16 rounds, fixed indices (fully unrolled).
#pragma unroll
    for (int round = 0; round < kM; ++round) {
        const int start = round & 1;
#pragma unroll
        for (int j = 0; j < kM - 1; j += 2) {
            const int a0 = j + start;
            if (a0 + 1 < kM) {
                if (d[a0] > d[a0 + 1]) {
                    float td = d[a0]; d[a0] = d[a0 + 1]; d[a0 + 1] = td;
                    float tp = p[a0]; p[a0] = p[a0 + 1]; p[a0 + 1] = tp;
                }
            }
        }
    }

    float cum = 1.0f, acc = 0.0f;
#pragma unroll
    for (int mm = 0; mm < kM; ++mm) {
        acc += d[mm] * p[mm] * cum;
        cum *= (1.0f - p[mm]);
    }

    red[threadIdx.x] = acc;
    __syncthreads();
    for (int stride = 128; stride > 0; stride >>= 1) {
        if (threadIdx.x < stride) red[threadIdx.x] += red[threadIdx.x + stride];
        __syncthreads();
    }
    if (threadIdx.x == 0) partials[blockIdx.x] = red[0];
}

// ---------------------------------------------------------------------------
// Kernel 3: prim_to_pcl (superquadric area weighting) + final combine.
// Single block of 256 threads = one thread per (b,m). Deterministic fixed-order
// final sums done by thread 0.
// ---------------------------------------------------------------------------
__global__ __launch_bounds__(256) void finalize_kernel(
    const float* __restrict__ d2,      // (B,M,S)
    const float* __restrict__ sizev,   // (B,M,3)
    const float* __restrict__ probs,   // (B,M)
    const float* __restrict__ partials,// (128)
    float* __restrict__ out)           // (4): total, pcl_to_prim, prim_to_pcl, regs
{
    __shared__ float areas[kB * kM];
    __shared__ float red[kB * kM];
    const int tid = threadIdx.x;       // 0..255
    const int b = tid / kM;
    const int m = tid % kM;

    const float s0 = sizev[(b * kM + m) * 3 + 0];
    const float s1 = sizev[(b * kM + m) * 3 + 1];
    const float s2 = sizev[(b * kM + m) * 3 + 2];
    const float area = kFourPi * powf(powf(s0 * s1, 1.6f) / 3.0f +
                                      powf(s0 * s2, 1.6f) / 3.0f +
                                      powf(s1 * s2, 1.6f) / 3.0f, 0.625f);
    areas[tid] = area;
    __syncthreads();

    float asum = 0.0f;
#pragma unroll
    for (int mm = 0; mm < kM; ++mm) asum += areas[b * kM + mm];
    const float wgt = (float)kM * area / asum;

    float meand2 = 0.0f;
    for (int s = 0; s < kS; ++s) {
        const float v = d2[(size_t)tid * kS + s];
        meand2 += (v >= kInfClamp) ? 0.0f : v;
    }
    meand2 *= (1.0f / (float)kS);

    red[tid] = meand2 * probs[b * kM + m] * wgt;
    __syncthreads();

    if (tid == 0) {
        float p2p = 0.0f;
        for (int i = 0; i < kB * kM; ++i) p2p += red[i];
        p2p /= (float)(kB * kM);
        float pcp = 0.0f;
        for (int i = 0; i < (kB * kN) / 256; ++i) pcp += partials[i];
        pcp /= (float)(kB * kN);
        out[1] = pcp;
        out[2] = p2p;
        out[3] = 0.0f;
        out[0] = pcp + p2p;
    }
}

// ---------------------------------------------------------------------------
extern "C" void kernel_launch(void* const* d_in, const int* in_sizes, int n_in,
                              void* d_out, int out_size, void* d_ws, size_t ws_size,
                              hipStream_t stream) {
    const float* pcl   = (const float*)d_in[0];   // (B,N,M,3)
    const float* prim  = (const float*)d_in[1];   // (B,M,S,3)
    const float* sizev = (const float*)d_in[2];   // (B,M,3)
    const float* probs = (const float*)d_in[3];   // (B,M)
    float* out = (float*)d_out;

    float* d1       = (float*)d_ws;                         // B*N*M  = 524288 f
    float* d2       = d1 + (size_t)kB * kN * kM;            // B*M*S  = 32768 f
    float* partials = d2 + (size_t)kB * kM * kS;            // 128 f

    dist_min_kernel<<<kB * kM, 256, 0, stream>>>(pcl, prim, d1, d2);
    pcl_to_prim_kernel<<<(kB * kN) / 256, 256, 0, stream>>>(d1, probs, partials);
    finalize_kernel<<<1, 256, 0, stream>>>(d2, sizev, probs, partials, out);
}